// Deform_25812753449863
// MI455X (gfx1250) — compile-verified
//
#include <hip/hip_runtime.h>
#include <math.h>

typedef __attribute__((ext_vector_type(2))) float f32x2;
typedef __attribute__((ext_vector_type(8))) float f32x8;
typedef __attribute__((ext_vector_type(4))) unsigned int u32x4;
typedef __attribute__((ext_vector_type(4))) int i32x4;
typedef __attribute__((ext_vector_type(8))) int i32x8;

#define PI_F 3.14159265358979323846f

#if defined(__has_builtin)
#if __has_builtin(__builtin_amdgcn_tensor_load_to_lds)
#define HAVE_TDM 1
#endif
#endif

// ---------------- workspace layout (float offsets) ----------------
// F buffers padded to stride 48 so each image's spectrum is one contiguous
// 3072-DWORD region == the [64][48] LDS tile (1-D TDM descriptor, zero pad free).
static constexpr size_t T_DR   = 0;                       // [64][64] forward cos
static constexpr size_t T_DI   = 4096;                    // [64][64] forward sin
static constexpr size_t T_ER   = 8192;                    // [64][64] inverse cos /64
static constexpr size_t T_EI   = 12288;                   // [64][64] inverse sin /64
static constexpr size_t T_E33R = 16384;                   // [48][36] inverse-33 cos /33 (zero padded)
static constexpr size_t T_E33I = 18112;                   // [48][36] inverse-33 sin /33
static constexpr size_t A_T    = 20480;                   // inT [B,C,56,56] = 25,690,112 f
static constexpr size_t A_M    = A_T;                     // mbuf [B,64,33,C] reuses inT (dead by then)
static constexpr size_t A_FR   = A_T + 25690112;          // Fr [BC,64,48]
static constexpr size_t A_FI   = A_FR + 25165824;         // Fi [BC,64,48]
static constexpr size_t A_S    = A_FI + 25165824;         // s   [B,C,64]
static constexpr size_t A_BOX  = A_S + 524288;            // box [B,C,64]
static constexpr size_t A_MSUM = A_BOX + 524288;          // msum [B,64,33]
static constexpr size_t A_PMIN = A_MSUM + 67584;          // 1024 partial mins
static constexpr size_t A_PMAX = A_PMIN + 1024;           // 1024 partial maxs
static constexpr size_t A_MM   = A_PMAX + 1024;           // mn,mx
static constexpr size_t A_XG   = A_MM + 2;                // [B,31,31]
static constexpr size_t A_YG   = A_XG + 30752;            // [B,31,31]

// ---------------- WMMA helper (CDNA5 f32 16x16x4) ----------------
__device__ __forceinline__ f32x8 wmma4(f32x2 a, f32x2 b, f32x8 c) {
  // (neg_a, A, neg_b, B, c_mod, C, reuse_a, reuse_b)
  return __builtin_amdgcn_wmma_f32_16x16x4_f32(false, a, false, b, (short)0, c,
                                               false, false);
}

__device__ __forceinline__ void vzero(f32x8& v) {
#pragma unroll
  for (int i = 0; i < 8; ++i) v[i] = 0.0f;
}

__device__ __forceinline__ float dnn(float a, float b) {  // divide_no_nan
  return (b == 0.0f) ? 0.0f : a / b;
}

// half-pixel bilinear source coords: src length -> dst length
__device__ __forceinline__ void rzc(int o, int src, int dst, int& i0, int& i1, float& w) {
  float f = ((float)o + 0.5f) * ((float)src / (float)dst) - 0.5f;
  f = fminf(fmaxf(f, 0.0f), (float)(src - 1));
  int a = (int)floorf(f);
  i0 = a;
  i1 = (a + 1 < src - 1) ? (a + 1) : (src - 1);
  w = f - (float)a;
}

#ifdef HAVE_TDM
// ---------------- Tensor Data Mover: contiguous 1-D DWORD tile -> LDS ----------
// D# per CDNA5 ISA (clang-23 6-arg builtin form).
__device__ __forceinline__ void tdm_load_1d(const void* gsrc, void* lds_dst,
                                            unsigned ndwords) {
  unsigned long long ga = (unsigned long long)(uintptr_t)gsrc;
  unsigned la = (unsigned)(uintptr_t)lds_dst;  // LDS aperture: addr[31:0] = offset
  u32x4 g0;
  g0[0] = 1u;                                       // count=1 (valid), user mode
  g0[1] = la;                                       // lds_addr [63:32]
  g0[2] = (unsigned)(ga & 0xFFFFFFFFull);           // global_addr lo
  g0[3] = (unsigned)((ga >> 32) & 0x01FFFFFFull)    // global_addr [56:32]
        | (2u << 30);                               // type=2 ("image")
  i32x8 g1;
  g1[0] = (int)(2u << 16);                          // data_size=2 (4 bytes)
  g1[1] = (int)((ndwords & 0xFFFFu) << 16);         // tensor_dim0 lo16 (bits 63:48)
  g1[2] = (int)((ndwords >> 16) | (1u << 16));      // tensor_dim0 hi16 | tensor_dim1=1
  g1[3] = (int)((ndwords & 0xFFFFu) << 16);         // tile_dim0 (bits 127:112)
  g1[4] = 0;                                        // tile_dim1=0, tile_dim2=0 (unused)
  g1[5] = (int)ndwords;                             // tensor_dim0_stride lo32
  g1[6] = 0;                                        // stride hi | dim1_stride lo
  g1[7] = 0;
  i32x4 z4;
  z4[0] = 0; z4[1] = 0; z4[2] = 0; z4[3] = 0;       // groups 2/3 unused (<=2D)
  i32x8 z8;
#pragma unroll
  for (int i = 0; i < 8; ++i) z8[i] = 0;
  __builtin_amdgcn_tensor_load_to_lds(g0, g1, z4, z4, z8, 0);
}
#endif

// ---------------- DFT matrix tables ----------------
__global__ void k_init(float* Dr, float* Di, float* Er, float* Ei,
                       float* E33r, float* E33i) {
  int i = blockIdx.x * 256 + threadIdx.x;
  if (i < 4096) {
    int m = i >> 6, k = i & 63;
    float a = -2.0f * PI_F * (float)((m * k) & 63) / 64.0f;  // forward sign
    Dr[i] = cosf(a);
    Di[i] = sinf(a);
    Er[i] = cosf(a) * (1.0f / 64.0f);    // cos even
    Ei[i] = -sinf(a) * (1.0f / 64.0f);   // inverse = +2pi phase, /64
  }
  if (i < 48 * 36) {
    int n = i / 36, k = i % 36;
    if (n < 33 && k < 33) {
      float a = 2.0f * PI_F * (float)((n * k) % 33) / 33.0f;
      E33r[i] = cosf(a) * (1.0f / 33.0f);
      E33i[i] = sinf(a) * (1.0f / 33.0f);
    } else {
      E33r[i] = 0.0f;  // zero pad keeps WMMA K/N tiles exact
      E33i[i] = 0.0f;
    }
  }
}

// ---------------- NHWC -> NCHW transpose of the input ----------------
__global__ void k_tr(const float* __restrict__ in, float* __restrict__ inT) {
  size_t i = (size_t)blockIdx.x * 256 + threadIdx.x;  // exactly 25,690,112 threads
  __builtin_prefetch(in + i + 65536, 0, 0);           // global_prefetch_b8 (speculative)
  int c = (int)(i & 255);
  size_t t = i >> 8;
  int x = (int)(t % 56); t /= 56;
  int y = (int)(t % 56);
  int b = (int)(t / 56);
  inT[(((size_t)b * 256 + c) * 56 + y) * 56 + x] = in[i];
}

// ---------------- forward: resize 56->64 + rfft2 (as f32 WMMA GEMMs) ----------------
__global__ __launch_bounds__(128) void k_fwd(const float* __restrict__ inT,
                                             const float* __restrict__ Dr,
                                             const float* __restrict__ Di,
                                             float* __restrict__ Frg,
                                             float* __restrict__ Fig,
                                             float* __restrict__ Sb) {
  int img = blockIdx.x;  // b*C + c
  __shared__ float Xs[64][64];
  __shared__ float Ar[64][64];  // Y then F (real)
  __shared__ float Ai[64][64];  // Y then F (imag)

  const float* plane = inT + (size_t)img * 3136;
  for (int p = threadIdx.x; p < 4096; p += 128) {
    int yy = p >> 6, xx = p & 63;
    int y0, y1, x0, x1; float wy, wx;
    rzc(yy, 56, 64, y0, y1, wy);
    rzc(xx, 56, 64, x0, x1, wx);
    float a = plane[y0 * 56 + x0], bq = plane[y0 * 56 + x1];
    float cq = plane[y1 * 56 + x0], dq = plane[y1 * 56 + x1];
    Xs[yy][xx] = a * (1 - wy) * (1 - wx) + bq * (1 - wy) * wx +
                 cq * wy * (1 - wx) + dq * wy * wx;
  }
  __syncthreads();

  int w = threadIdx.x >> 5, lane = threadIdx.x & 31;
  int hi = lane >> 4, lo = lane & 15;
  int m0 = w * 16;

  // ---- stage 1: Y = D @ X  (M=64 split over 4 waves, N=64, K=64) ----
  f32x8 aR[4], aI[4];
#pragma unroll
  for (int nt = 0; nt < 4; ++nt) { vzero(aR[nt]); vzero(aI[nt]); }
  for (int k0 = 0; k0 < 64; k0 += 4) {
    int arow = (m0 + lo) * 64 + k0 + 2 * hi;
    f32x2 ar, ai;
    ar.x = Dr[arow]; ar.y = Dr[arow + 1];
    ai.x = Di[arow]; ai.y = Di[arow + 1];
#pragma unroll
    for (int nt = 0; nt < 4; ++nt) {
      f32x2 b;
      b.x = Xs[k0 + 2 * hi][nt * 16 + lo];
      b.y = Xs[k0 + 2 * hi + 1][nt * 16 + lo];
      aR[nt] = wmma4(ar, b, aR[nt]);
      aI[nt] = wmma4(ai, b, aI[nt]);
    }
  }
#pragma unroll
  for (int nt = 0; nt < 4; ++nt)
#pragma unroll
    for (int v = 0; v < 8; ++v) {
      Ar[m0 + v + 8 * hi][nt * 16 + lo] = aR[nt][v];
      Ai[m0 + v + 8 * hi][nt * 16 + lo] = aI[nt][v];
    }
  __syncthreads();

  // ---- stage 2: F = Y @ D^T  (N=48 tiles, keep 33 rfft bins) ----
  f32x8 fR[3], fI[3];
#pragma unroll
  for (int nt = 0; nt < 3; ++nt) { vzero(fR[nt]); vzero(fI[nt]); }
  for (int k0 = 0; k0 < 64; k0 += 4) {
    f32x2 yr, yi;
    yr.x = Ar[m0 + lo][k0 + 2 * hi]; yr.y = Ar[m0 + lo][k0 + 2 * hi + 1];
    yi.x = Ai[m0 + lo][k0 + 2 * hi]; yi.y = Ai[m0 + lo][k0 + 2 * hi + 1];
#pragma unroll
    for (int nt = 0; nt < 3; ++nt) {
      int bro = (nt * 16 + lo) * 64 + k0 + 2 * hi;
      f32x2 br, bi, nbi;
      br.x = Dr[bro]; br.y = Dr[bro + 1];
      bi.x = Di[bro]; bi.y = Di[bro + 1];
      nbi.x = -bi.x;  nbi.y = -bi.y;
      fR[nt] = wmma4(yr, br, fR[nt]);
      fR[nt] = wmma4(yi, nbi, fR[nt]);   // - Yi*Di^T
      fI[nt] = wmma4(yr, bi, fI[nt]);
      fI[nt] = wmma4(yi, br, fI[nt]);
    }
  }
  __syncthreads();  // all waves done reading Y before overwrite
#pragma unroll
  for (int nt = 0; nt < 3; ++nt)
#pragma unroll
    for (int v = 0; v < 8; ++v) {
      Ar[m0 + v + 8 * hi][nt * 16 + lo] = fR[nt][v];
      Ai[m0 + v + 8 * hi][nt * 16 + lo] = fI[nt][v];
    }
  __syncthreads();

  // store stride-48 rows, cols >=33 zeroed (feeds zero-padded WMMA + TDM tile)
  size_t base = (size_t)img * 64 * 48;
  for (int p = threadIdx.x; p < 64 * 48; p += 128) {
    int r = p / 48, cc = p % 48;
    float vr = (cc < 33) ? Ar[r][cc] : 0.0f;
    float vi = (cc < 33) ? Ai[r][cc] : 0.0f;
    Frg[base + p] = vr;
    Fig[base + p] = vi;
  }
  if (threadIdx.x < 64) {  // s[b][c][ky] = sum_kx log|F|
    int r = threadIdx.x;
    float s = 0.0f;
    for (int cc = 0; cc < 33; ++cc) {
      float mr = Ar[r][cc], mi = Ai[r][cc];
      float mag = sqrtf(mr * mr + mi * mi);
      s += (mag == 0.0f) ? 0.0f : logf(mag);
    }
    Sb[(size_t)img * 64 + r] = s;
  }
}

// ---------------- 3x3 box over the (C, ky) plane, zero padded ----------------
__global__ void k_box(const float* __restrict__ S, float* __restrict__ Box) {
  int b = blockIdx.x;       // 32
  int c = threadIdx.x;      // 256
  const float* Sp = S + (size_t)b * 256 * 64;
  float* Bp = Box + (size_t)b * 256 * 64;
  for (int ky = 0; ky < 64; ++ky) {
    float s = 0.0f;
    for (int dc = -1; dc <= 1; ++dc) {
      int cc = c + dc;
      if (cc < 0 || cc >= 256) continue;
      for (int dk = -1; dk <= 1; ++dk) {
        int kk = ky + dk;
        if (kk < 0 || kk >= 64) continue;
        s += Sp[cc * 64 + kk];
      }
    }
    Bp[c * 64 + ky] = s;
  }
}

// ---------------- spectral residual ratio, in place on F ----------------
__global__ void k_ratio(float* __restrict__ Fr, float* __restrict__ Fi,
                        const float* __restrict__ Box) {
  size_t i = (size_t)blockIdx.x * 256 + threadIdx.x;
  if (i >= (size_t)8192 * 64 * 33) return;
  int cc = (int)(i % 33);
  size_t t = i / 33;
  int r = (int)(t % 64);
  size_t img = t / 64;
  size_t off = img * 64 * 48 + (size_t)r * 48 + cc;
  float mr = Fr[off], mi = Fi[off];
  float mag = sqrtf(mr * mr + mi * mi);
  float magw = (mag == 0.0f) ? 1.0f : mag;
  float logm = (mag == 0.0f) ? 0.0f : logf(mag);
  float sr = logm - Box[img * 64 + r];
  float ratio = (mag == 0.0f) ? 0.0f : (sr / magw);
  Fr[off] = mr * ratio;
  Fi[off] = mi * ratio;
}

// ---------------- inverse: full complex ifft2 on [64,33] + |.|^2 ----------------
__global__ __launch_bounds__(128) void k_inv(const float* __restrict__ Frg,
                                             const float* __restrict__ Fig,
                                             const float* __restrict__ Er,
                                             const float* __restrict__ Ei,
                                             const float* __restrict__ E33r,
                                             const float* __restrict__ E33i,
                                             float* __restrict__ mbuf) {
  int img = blockIdx.x;
  int b = img >> 8, c = img & 255;
  __shared__ float Crs[64][48];
  __shared__ float Cis[64][48];
  __shared__ float Trs[64][48];
  __shared__ float Tis[64][48];

  size_t base = (size_t)img * 64 * 48;
#ifdef HAVE_TDM
  // Tensor Data Mover DMA: one contiguous 3072-DWORD tile per buffer,
  // issued by wave 0; waves idle on the cheap TENSORcnt instead of VMEM+DS.
  if ((threadIdx.x >> 5) == 0) {
    tdm_load_1d(Frg + base, &Crs[0][0], 64 * 48);
    tdm_load_1d(Fig + base, &Cis[0][0], 64 * 48);
    __builtin_amdgcn_s_wait_tensorcnt(0);
  }
#else
  for (int p = threadIdx.x; p < 64 * 48; p += 128) {
    Crs[0][p] = Frg[base + p];
    Cis[0][p] = Fig[base + p];
  }
#endif
  __syncthreads();

  int w = threadIdx.x >> 5, lane = threadIdx.x & 31;
  int hi = lane >> 4, lo = lane & 15;
  int m0 = w * 16;

  // ---- stage 1: T = E64 @ C  (complex), N=48 tiles ----
  f32x8 tR[3], tI[3];
#pragma unroll
  for (int nt = 0; nt < 3; ++nt) { vzero(tR[nt]); vzero(tI[nt]); }
  for (int k0 = 0; k0 < 64; k0 += 4) {
    int arow = (m0 + lo) * 64 + k0 + 2 * hi;
    f32x2 er, ei, nei;
    er.x = Er[arow]; er.y = Er[arow + 1];
    ei.x = Ei[arow]; ei.y = Ei[arow + 1];
    nei.x = -ei.x;   nei.y = -ei.y;
#pragma unroll
    for (int nt = 0; nt < 3; ++nt) {
      f32x2 br, bi;
      br.x = Crs[k0 + 2 * hi][nt * 16 + lo];
      br.y = Crs[k0 + 2 * hi + 1][nt * 16 + lo];
      bi.x = Cis[k0 + 2 * hi][nt * 16 + lo];
      bi.y = Cis[k0 + 2 * hi + 1][nt * 16 + lo];
      tR[nt] = wmma4(er, br, tR[nt]);
      tR[nt] = wmma4(nei, bi, tR[nt]);
      tI[nt] = wmma4(er, bi, tI[nt]);
      tI[nt] = wmma4(ei, br, tI[nt]);
    }
  }
  __syncthreads();
#pragma unroll
  for (int nt = 0; nt < 3; ++nt)
#pragma unroll
    for (int v = 0; v < 8; ++v) {
      Trs[m0 + v + 8 * hi][nt * 16 + lo] = tR[nt][v];
      Tis[m0 + v + 8 * hi][nt * 16 + lo] = tI[nt][v];
    }
  __syncthreads();

  // ---- stage 2: O = T @ E33^T  (K padded to 36, N=48 tiles, keep 33) ----
  f32x8 oR[3], oI[3];
#pragma unroll
  for (int nt = 0; nt < 3; ++nt) { vzero(oR[nt]); vzero(oI[nt]); }
  for (int k0 = 0; k0 < 36; k0 += 4) {
    f32x2 tr, ti;
    tr.x = Trs[m0 + lo][k0 + 2 * hi]; tr.y = Trs[m0 + lo][k0 + 2 * hi + 1];
    ti.x = Tis[m0 + lo][k0 + 2 * hi]; ti.y = Tis[m0 + lo][k0 + 2 * hi + 1];
#pragma unroll
    for (int nt = 0; nt < 3; ++nt) {
      int bro = (nt * 16 + lo) * 36 + k0 + 2 * hi;
      f32x2 br, bi, nbi;
      br.x = E33r[bro]; br.y = E33r[bro + 1];
      bi.x = E33i[bro]; bi.y = E33i[bro + 1];
      nbi.x = -bi.x;    nbi.y = -bi.y;
      oR[nt] = wmma4(tr, br, oR[nt]);
      oR[nt] = wmma4(ti, nbi, oR[nt]);
      oI[nt] = wmma4(tr, bi, oI[nt]);
      oI[nt] = wmma4(ti, br, oI[nt]);
    }
  }

  // m[b, y, x, c] = |O|^2   (NHWC-of-(64,33) for the downstream resizes)
#pragma unroll
  for (int nt = 0; nt < 3; ++nt)
#pragma unroll
    for (int v = 0; v < 8; ++v) {
      int row = m0 + v + 8 * hi, col = nt * 16 + lo;
      if (col < 33) {
        float vr = oR[nt][v], vi = oI[nt][v];
        mbuf[(((size_t)b * 64 + row) * 33 + col) * 256 + c] = vr * vr + vi * vi;
      }
    }
}

// ---------------- channel sum of m ----------------
__global__ void k_msum(const float* __restrict__ mbuf, float* __restrict__ msum) {
  int o = blockIdx.x;  // (b*64+y)*33+x, 67584 blocks
  __shared__ float red[256];
  red[threadIdx.x] = mbuf[(size_t)o * 256 + threadIdx.x];
  __syncthreads();
  for (int s = 128; s > 0; s >>= 1) {
    if (threadIdx.x < s) red[threadIdx.x] += red[threadIdx.x + s];
    __syncthreads();
  }
  if (threadIdx.x == 0) msum[o] = red[0];
}

// ---------------- min/max over the RESIZED per-channel m (never stored) ----------------
__global__ void k_minmax(const float* __restrict__ mbuf,
                         float* __restrict__ pmin, float* __restrict__ pmax) {
  const size_t N = (size_t)32 * 56 * 56 * 256;
  float mn = 3.4e38f, mx = -3.4e38f;
  for (size_t i = (size_t)blockIdx.x * 256 + threadIdx.x; i < N;
       i += (size_t)gridDim.x * 256) {
    int c = (int)(i & 255);
    size_t t = i >> 8;
    int x = (int)(t % 56); t /= 56;
    int y = (int)(t % 56);
    int b = (int)(t / 56);
    int y0, y1, x0, x1; float wy, wx;
    rzc(y, 64, 56, y0, y1, wy);
    rzc(x, 33, 56, x0, x1, wx);
    const float* base = mbuf + (size_t)b * 64 * 33 * 256;
    float a = base[((size_t)y0 * 33 + x0) * 256 + c];
    float bq = base[((size_t)y0 * 33 + x1) * 256 + c];
    float cq = base[((size_t)y1 * 33 + x0) * 256 + c];
    float dq = base[((size_t)y1 * 33 + x1) * 256 + c];
    float v = a * (1 - wy) * (1 - wx) + bq * (1 - wy) * wx +
              cq * wy * (1 - wx) + dq * wy * wx;
    mn = fminf(mn, v);
    mx = fmaxf(mx, v);
  }
  __shared__ float rmn[256], rmx[256];
  rmn[threadIdx.x] = mn;
  rmx[threadIdx.x] = mx;
  __syncthreads();
  for (int s = 128; s > 0; s >>= 1) {
    if (threadIdx.x < s) {
      rmn[threadIdx.x] = fminf(rmn[threadIdx.x], rmn[threadIdx.x + s]);
      rmx[threadIdx.x] = fmaxf(rmx[threadIdx.x], rmx[threadIdx.x + s]);
    }
    __syncthreads();
  }
  if (threadIdx.x == 0) {
    pmin[blockIdx.x] = rmn[0];
    pmax[blockIdx.x] = rmx[0];
  }
}

__global__ void k_mmfin(const float* __restrict__ pmin, const float* __restrict__ pmax,
                        float* __restrict__ mm) {
  __shared__ float rmn[256], rmx[256];
  float mn = 3.4e38f, mx = -3.4e38f;
  for (int i = threadIdx.x; i < 1024; i += 256) {
    mn = fminf(mn, pmin[i]);
    mx = fmaxf(mx, pmax[i]);
  }
  rmn[threadIdx.x] = mn;
  rmx[threadIdx.x] = mx;
  __syncthreads();
  for (int s = 128; s > 0; s >>= 1) {
    if (threadIdx.x < s) {
      rmn[threadIdx.x] = fminf(rmn[threadIdx.x], rmn[threadIdx.x + s]);
      rmx[threadIdx.x] = fmaxf(rmx[threadIdx.x], rmx[threadIdx.x + s]);
    }
    __syncthreads();
  }
  if (threadIdx.x == 0) {
    mm[0] = rmn[0];
    mm[1] = rmx[0];
  }
}

// ---------------- deformation grid: resize->pad->gaussian conv ----------------
__global__ __launch_bounds__(256) void k_grid(const float* __restrict__ msum,
                                              const float* __restrict__ mm,
                                              float* __restrict__ xg31,
                                              float* __restrict__ yg31) {
  int b = blockIdx.x;  // 32 blocks
  __shared__ float S56s[56][56];  // sum_c sal, 56x56
  __shared__ float s91[91][92];   // reflect-padded 31x31 resize
  __shared__ float gk[61][61];    // gaussian fwhm=13
  float mn = mm[0], mx = mm[1];
  float d = mx - mn;
  float invd = (d == 0.0f) ? 0.0f : 1.0f / d;
  const float* mp = msum + (size_t)b * 64 * 33;

  for (int p = threadIdx.x; p < 56 * 56; p += 256) {
    int y = p / 56, x = p % 56;
    int y0, y1, x0, x1; float wy, wx;
    rzc(y, 64, 56, y0, y1, wy);
    rzc(x, 33, 56, x0, x1, wx);
    float a = mp[y0 * 33 + x0], bq = mp[y0 * 33 + x1];
    float cq = mp[y1 * 33 + x0], dq = mp[y1 * 33 + x1];
    float v = a * (1 - wy) * (1 - wx) + bq * (1 - wy) * wx +
              cq * wy * (1 - wx) + dq * wy * wx;
    S56s[y][x] = (v - 256.0f * mn) * invd;  // sum_c of normalized sal
  }
  for (int p = threadIdx.x; p < 61 * 61; p += 256) {
    int i = p / 61, j = p % 61;
    float r2 = (float)((i - 30) * (i - 30) + (j - 30) * (j - 30));
    gk[i][j] = expf(-4.0f * 0.69314718f * r2 / 169.0f);
  }
  __syncthreads();
  for (int p = threadIdx.x; p < 91 * 91; p += 256) {
    int i = p / 91, j = p % 91;
    int ri = i - 30; if (ri < 0) ri = -ri; if (ri > 30) ri = 60 - ri;  // reflect
    int rj = j - 30; if (rj < 0) rj = -rj; if (rj > 30) rj = 60 - rj;
    int y0, y1, x0, x1; float wy, wx;
    rzc(ri, 56, 31, y0, y1, wy);
    rzc(rj, 56, 31, x0, x1, wx);
    float a = S56s[y0][x0], bq = S56s[y0][x1];
    float cq = S56s[y1][x0], dq = S56s[y1][x1];
    s91[i][j] = a * (1 - wy) * (1 - wx) + bq * (1 - wy) * wx +
                cq * wy * (1 - wx) + dq * wy * wx;
  }
  __syncthreads();
  for (int o = threadIdx.x; o < 31 * 31; o += 256) {
    int oy = o / 31, ox = o % 31;
    float p = 0.0f, mxa = 0.0f, mya = 0.0f;
    for (int i = 0; i < 61; ++i) {
      float cy = ((float)(oy + i) - 30.0f) * (1.0f / 30.0f);
      for (int j = 0; j < 61; ++j) {
        float gs_ = gk[i][j] * s91[oy + i][ox + j];
        float cx = ((float)(ox + j) - 30.0f) * (1.0f / 30.0f);
        p += gs_;
        mxa += gs_ * cx;
        mya += gs_ * cy;
      }
    }
    float xv = fminf(fmaxf(2.0f * dnn(mxa, p) - 1.0f, -1.0f), 1.0f);
    float yv = fminf(fmaxf(2.0f * dnn(mya, p) - 1.0f, -1.0f), 1.0f);
    xg31[(size_t)b * 961 + o] = xv;
    yg31[(size_t)b * 961 + o] = yv;
  }
}

// ---------------- final: resize grid 31->56 + bilinear grid sample ----------------
__global__ __launch_bounds__(256) void k_sample(const float* __restrict__ in,
                                                const float* __restrict__ xg31,
                                                const float* __restrict__ yg31,
                                                float* __restrict__ out) {
  int blk = blockIdx.x;  // b*3136 + y*56 + x, 100352 blocks
  int x = blk % 56;
  int t = blk / 56;
  int y = t % 56;
  int b = t / 56;
  int y0, y1, x0, x1; float wy, wx;
  rzc(y, 31, 56, y0, y1, wy);
  rzc(x, 31, 56, x0, x1, wx);
  const float* xp = xg31 + (size_t)b * 961;
  const float* yp = yg31 + (size_t)b * 961;
  float w00 = (1 - wy) * (1 - wx), w01 = (1 - wy) * wx;
  float w10 = wy * (1 - wx), w11 = wy * wx;
  float xgv = xp[y0 * 31 + x0] * w00 + xp[y0 * 31 + x1] * w01 +
              xp[y1 * 31 + x0] * w10 + xp[y1 * 31 + x1] * w11;
  float ygv = yp[y0 * 31 + x0] * w00 + yp[y0 * 31 + x1] * w01 +
              yp[y1 * 31 + x0] * w10 + yp[y1 * 31 + x1] * w11;
  float fx = 0.5f * ((xgv + 1.0f) * 54.0f);  // (max_x - 1) = 54
  float fy = 0.5f * ((ygv + 1.0f) * 54.0f);
  int fx0 = (int)floorf(fx);
  int fy0 = (int)floorf(fy);
  int sx0 = min(max(fx0, 0), 55), sx1 = min(max(fx0 + 1, 0), 55);
  int sy0 = min(max(fy0, 0), 55), sy1 = min(max(fy0 + 1, 0), 55);
  float wa = ((float)sx1 - fx) * ((float)sy1 - fy);
  float wb = ((float)sx1 - fx) * (fy - (float)sy0);
  float wc = (fx - (float)sx0) * ((float)sy1 - fy);
  float wd = (fx - (float)sx0) * (fy - (float)sy0);
  int c = threadIdx.x;
  size_t base = (size_t)b * 3136 * 256;
  float Ia = in[base + ((size_t)sy0 * 56 + sx0) * 256 + c];
  float Ib = in[base + ((size_t)sy1 * 56 + sx0) * 256 + c];
  float Ic = in[base + ((size_t)sy0 * 56 + sx1) * 256 + c];
  float Id = in[base + ((size_t)sy1 * 56 + sx1) * 256 + c];
  out[(size_t)blk * 256 + c] = wa * Ia + wb * Ib + wc * Ic + wd * Id;
}

// ---------------- launch ----------------
extern "C" void kernel_launch(void* const* d_in, const int* in_sizes, int n_in,
                              void* d_out, int out_size, void* d_ws, size_t ws_size,
                              hipStream_t stream) {
  (void)in_sizes; (void)n_in; (void)out_size; (void)ws_size;
  const float* in = (const float*)d_in[0];
  float* out = (float*)d_out;
  float* ws = (float*)d_ws;

  float* Dr = ws + T_DR;   float* Di = ws + T_DI;
  float* Er = ws + T_ER;   float* Ei = ws + T_EI;
  float* E33r = ws + T_E33R; float* E33i = ws + T_E33I;
  float* inT = ws + A_T;
  float* mbuf = ws + A_M;   // reuses inT region (inT dead after k_fwd)
  float* Fr = ws + A_FR;   float* Fi = ws + A_FI;
  float* S = ws + A_S;     float* Box = ws + A_BOX;
  float* msum = ws + A_MSUM;
  float* pmin = ws + A_PMIN; float* pmax = ws + A_PMAX;
  float* mm = ws + A_MM;
  float* xg31 = ws + A_XG; float* yg31 = ws + A_YG;

  k_init<<<16, 256, 0, stream>>>(Dr, Di, Er, Ei, E33r, E33i);
  k_tr<<<100352, 256, 0, stream>>>(in, inT);
  k_fwd<<<8192, 128, 0, stream>>>(inT, Dr, Di, Fr, Fi, S);
  k_box<<<32, 256, 0, stream>>>(S, Box);
  k_ratio<<<67584, 256, 0, stream>>>(Fr, Fi, Box);
  k_inv<<<8192, 128, 0, stream>>>(Fr, Fi, Er, Ei, E33r, E33i, mbuf);
  k_msum<<<67584, 256, 0, stream>>>(mbuf, msum);
  k_minmax<<<1024, 256, 0, stream>>>(mbuf, pmin, pmax);
  k_mmfin<<<1, 256, 0, stream>>>(pmin, pmax, mm);
  k_grid<<<32, 256, 0, stream>>>(msum, mm, xg31, yg31);
  k_sample<<<100352, 256, 0, stream>>>(in, xg31, yg31, out);
}